// GraphSageConv_5574867550378
// MI455X (gfx1250) — compile-verified
//
#include <hip/hip_runtime.h>
#include <hip/hip_bf16.h>

typedef __attribute__((ext_vector_type(2))) float v2f;
typedef __attribute__((ext_vector_type(8))) float v8f;

#define FDIM   128
#define UNITS  128
#define OUTW   (2 * UNITS)
#define KEEP_THRESH 3006477107u   /* 0.7 * 2^32 */
#define INV_KEEP    (1.0f / 0.7f)

__device__ __forceinline__ unsigned wang_hash(unsigned s) {
    s = (s ^ 61u) ^ (s >> 16);
    s *= 9u;
    s ^= s >> 4;
    s *= 0x27d4eb2du;
    s ^= s >> 15;
    return s;
}

// ---------------- zero workspace (seg_sum + counts) ----------------
__global__ void zero_kernel(float* __restrict__ p, int n) {
    int i = blockIdx.x * blockDim.x + threadIdx.x;
    if (i < n) p[i] = 0.0f;
}

// ---------------- pack W into per-lane WMMA fragment order ----------------
// P[((j*32 + kstep)*32 + lane)*2 + v] = W[(4*kstep + 2*(lane>>4) + v)*UNITS + j*16 + (lane&15)]
__global__ void pack_w_kernel(const float* __restrict__ Ws, const float* __restrict__ Wn,
                              float* __restrict__ Ps, float* __restrict__ Pn) {
    int tid = blockIdx.x * blockDim.x + threadIdx.x;   // 0 .. 16383
    if (tid >= 2 * 8192) return;
    const float* W = (tid < 8192) ? Ws : Wn;
    float*       P = (tid < 8192) ? Ps : Pn;
    int t  = tid & 8191;
    int j  = t >> 10;          // column tile 0..7
    int kk = (t >> 5) & 31;    // k-step 0..31
    int l  = t & 31;           // lane
    int n  = l & 15;
    int k  = 4 * kk + ((l >> 4) << 1);
    P[2 * t + 0] = W[(k + 0) * UNITS + j * 16 + n];
    P[2 * t + 1] = W[(k + 1) * UNITS + j * 16 + n];
}

// ---------------- edge scatter: wave per edge, f32 atomics into L2 ----------------
__global__ __launch_bounds__(256) void scatter_kernel(const float* __restrict__ x,
                                                      const long long* __restrict__ src,
                                                      const long long* __restrict__ dst,
                                                      float* __restrict__ seg_sum,
                                                      float* __restrict__ counts, int E) {
    int e = blockIdx.x * 8 + (threadIdx.x >> 5);
    if (e >= E) return;                       // wave-uniform
    int lane = threadIdx.x & 31;
    int s = (int)src[e];
    int d = (int)dst[e];
    const float* xr = x + (size_t)s * FDIM;
    float*       o  = seg_sum + (size_t)d * FDIM;
#pragma unroll
    for (int i = 0; i < 4; ++i)
        atomicAdd(o + lane + 32 * i, xr[lane + 32 * i]);
    if (lane == 0) atomicAdd(counts + d, 1.0f);
}

// ---------------- fused dual GEMM (WMMA f32 16x16x4) + bias + ELU + dropout ----------------
__global__ __launch_bounds__(256) void gemm_kernel(const float* __restrict__ x,
                                                   const float* __restrict__ seg_sum,
                                                   const float* __restrict__ counts,
                                                   const float* __restrict__ wpack_self,
                                                   const float* __restrict__ wpack_neigh,
                                                   const float* __restrict__ bias,
                                                   float* __restrict__ out, int N) {
    const int wave = threadIdx.x >> 5;
    const int lane = threadIdx.x & 31;
    const int row0 = (blockIdx.x * 8 + wave) * 16;
    if (row0 >= N) return;                    // wave-uniform: EXEC stays all-ones

    const int mloc = lane & 15;               // A-fragment row within tile
    const int m    = row0 + mloc;
    const int off  = (lane >> 4) << 1;        // k sub-offset 0 or 2
    const int n    = lane & 15;               // output column within tile
    const int half = lane >> 4;

#pragma unroll
    for (int p = 0; p < 2; ++p) {
        const float* Amat = p ? seg_sum : x;
        float scale = 1.0f;
        if (p) scale = 1.0f / fmaxf(counts[m], 1.0f);   // fold mean-division into A load

        // Load this wave's entire A k-sweep into registers: 32 steps x 2 floats/lane.
        const v2f* av = (const v2f*)(Amat + (size_t)m * FDIM + off);
        v2f a[32];
#pragma unroll
        for (int s = 0; s < 32; ++s) a[s] = av[2 * s] * scale;

        const float* wp = p ? wpack_neigh : wpack_self;

        for (int j = 0; j < 8; ++j) {
            const v2f* bv = (const v2f*)wp + (j * 32 * 32 + lane);
            v8f acc = {};
#pragma unroll
            for (int s = 0; s < 32; ++s) {
                v2f b = bv[s * 32];
                acc = __builtin_amdgcn_wmma_f32_16x16x4_f32(
                    false, a[s], false, b, (short)0, acc, false, false);
            }
            // epilogue: bias + ELU + deterministic dropout, store tile
            const int col = p * UNITS + j * 16 + n;
            const float bcol = bias[col];
#pragma unroll
            for (int r = 0; r < 8; ++r) {
                int row = row0 + r + 8 * half;
                float v = acc[r] + bcol;
                v = (v > 0.0f) ? v : expm1f(v);
                unsigned h = wang_hash(((unsigned)row * OUTW + (unsigned)col) ^ 0x9E3779B9u);
                v = (h < KEEP_THRESH) ? v * INV_KEEP : 0.0f;
                out[(size_t)row * OUTW + col] = v;
            }
        }
    }
}

// ---------------- row L2 normalize: one wave per 256-wide row ----------------
__global__ __launch_bounds__(256) void l2norm_kernel(float* __restrict__ out, int N) {
    int row  = blockIdx.x * 8 + (threadIdx.x >> 5);
    if (row >= N) return;                     // wave-uniform
    int lane = threadIdx.x & 31;
    float4* p = (float4*)(out + (size_t)row * OUTW);
    float4 v0 = p[2 * lane + 0];
    float4 v1 = p[2 * lane + 1];
    float s = v0.x * v0.x + v0.y * v0.y + v0.z * v0.z + v0.w * v0.w +
              v1.x * v1.x + v1.y * v1.y + v1.z * v1.z + v1.w * v1.w;
#pragma unroll
    for (int d = 16; d > 0; d >>= 1) s += __shfl_xor(s, d, 32);
    float r = rsqrtf(fmaxf(s, 1e-12f));
    v0.x *= r; v0.y *= r; v0.z *= r; v0.w *= r;
    v1.x *= r; v1.y *= r; v1.z *= r; v1.w *= r;
    p[2 * lane + 0] = v0;
    p[2 * lane + 1] = v1;
}

extern "C" void kernel_launch(void* const* d_in, const int* in_sizes, int n_in,
                              void* d_out, int out_size, void* d_ws, size_t ws_size,
                              hipStream_t stream) {
    const float*     x      = (const float*)d_in[0];
    const float*     W_self = (const float*)d_in[1];
    const float*     W_neigh= (const float*)d_in[2];
    const float*     bias   = (const float*)d_in[3];
    const long long* src    = (const long long*)d_in[4];
    const long long* dst    = (const long long*)d_in[5];
    float*           out    = (float*)d_out;

    const int N = in_sizes[0] / FDIM;
    const int E = in_sizes[4];

    // workspace layout
    float* seg_sum     = (float*)d_ws;                         // N*FDIM
    float* counts      = seg_sum + (size_t)N * FDIM;           // N
    float* wpack_self  = counts + N;                           // 16384
    float* wpack_neigh = wpack_self + 16384;                   // 16384

    int nz = N * FDIM + N;
    zero_kernel<<<(nz + 255) / 256, 256, 0, stream>>>(seg_sum, nz);
    pack_w_kernel<<<(2 * 8192 + 255) / 256, 256, 0, stream>>>(W_self, W_neigh,
                                                              wpack_self, wpack_neigh);
    scatter_kernel<<<(E + 7) / 8, 256, 0, stream>>>(x, src, dst, seg_sum, counts, E);

    int nwaves = (N + 15) / 16;
    gemm_kernel<<<(nwaves + 7) / 8, 256, 0, stream>>>(x, seg_sum, counts,
                                                      wpack_self, wpack_neigh,
                                                      bias, out, N);
    l2norm_kernel<<<(N + 7) / 8, 256, 0, stream>>>(out, N);
}